// MultiStepLIFNode_23124103921697
// MI455X (gfx1250) — compile-verified
//
#include <hip/hip_runtime.h>

// LIF sequential scan over T=16:  mem = 0.7*mem + 0.7*x; spike = (mem > 1); mem -= spike
// Memory-bound (256 MB traffic, ~134 MFLOP). Strategy:
//   - async global->LDS staging (gfx1250 GLOBAL_LOAD_ASYNC_TO_LDS_B128, ASYNCcnt)
//   - incremental s_wait_asynccnt drain (loads complete in order)
//   - nontemporal stores so the write-once output doesn't evict the (L2-resident) input
typedef float v4f __attribute__((ext_vector_type(4)));

#define T_STEPS 16
#define TPB 256
#define VEC 4
#define TILE_FLOATS (TPB * VEC)          // 1024 floats per block per timestep
#define TILE_BYTES  (TILE_FLOATS * 4)    // 4096 B

__global__ __launch_bounds__(TPB) void lif_scan_async(
    const float* __restrict__ x, float* __restrict__ out, int n_spatial)
{
    __shared__ v4f tile[T_STEPS][TPB];   // 64 KB: 5 workgroups / 320KB WGP -> 40 waves/WGP

    const int tid   = threadIdx.x;
    const int gbase = blockIdx.x * TILE_FLOATS + tid * VEC;

    if (gbase + VEC <= n_spatial) {
        const unsigned t_stride = (unsigned)n_spatial * 4u;        // bytes between timesteps
        const unsigned goff     = (unsigned)gbase * 4u;            // byte offset within a timestep
        const unsigned lds0     = (unsigned)(size_t)&tile[0][tid]; // low 32 bits == LDS byte offset
        const unsigned long long xbase = (unsigned long long)(size_t)x;

        // Issue all 16 per-lane 16B async copies back-to-back (ASYNCcnt -> 16).
        // saddr form: mem_addr = SGPR64 + VGPR32 offset (max offset ~128 MB < 2^31).
        #pragma unroll
        for (int t = 0; t < T_STEPS; ++t) {
            unsigned go = goff + (unsigned)t * t_stride;
            unsigned lo = lds0 + (unsigned)t * (unsigned)TILE_BYTES;
            asm volatile("global_load_async_to_lds_b128 %0, %1, %2"
                         :: "v"(lo), "v"(go), "s"(xbase)
                         : "memory");
        }

        v4f mem = {0.0f, 0.0f, 0.0f, 0.0f};
        v4f sp;

        // Async loads complete in order: step t only needs ASYNCcnt <= 15-t,
        // so compute/stores overlap the remaining copies.
        #define LIF_STEP(t, cnt)                                              \
        do {                                                                  \
            asm volatile("s_wait_asynccnt " #cnt ::: "memory");               \
            v4f xv = tile[t][tid];                                            \
            _Pragma("unroll")                                                 \
            for (int i = 0; i < 4; ++i) {                                     \
                mem[i] = 0.7f * mem[i] + 0.7f * xv[i];                        \
                float s = (mem[i] > 1.0f) ? 1.0f : 0.0f;                      \
                sp[i]  = s;                                                   \
                mem[i] -= s;                                                  \
            }                                                                 \
            __builtin_nontemporal_store(sp,                                   \
                (v4f*)(out + (size_t)(t) * (size_t)n_spatial + gbase));       \
        } while (0)

        LIF_STEP(0, 15);  LIF_STEP(1, 14);  LIF_STEP(2, 13);  LIF_STEP(3, 12);
        LIF_STEP(4, 11);  LIF_STEP(5, 10);  LIF_STEP(6, 9);   LIF_STEP(7, 8);
        LIF_STEP(8, 7);   LIF_STEP(9, 6);   LIF_STEP(10, 5);  LIF_STEP(11, 4);
        LIF_STEP(12, 3);  LIF_STEP(13, 2);  LIF_STEP(14, 1);  LIF_STEP(15, 0);
        #undef LIF_STEP
    } else {
        // Scalar tail (unused for the reference shape: 2M % 1024 == 0)
        for (int j = 0; j < VEC; ++j) {
            int idx = gbase + j;
            if (idx < n_spatial) {
                float mem = 0.0f;
                for (int t = 0; t < T_STEPS; ++t) {
                    mem = 0.7f * mem + 0.7f * x[(size_t)t * n_spatial + idx];
                    float s = (mem > 1.0f) ? 1.0f : 0.0f;
                    out[(size_t)t * n_spatial + idx] = s;
                    mem -= s;
                }
            }
        }
    }
}

extern "C" void kernel_launch(void* const* d_in, const int* in_sizes, int n_in,
                              void* d_out, int out_size, void* d_ws, size_t ws_size,
                              hipStream_t stream)
{
    const float* x  = (const float*)d_in[0];   // [16, 32, 64, 32, 32] f32
    float*      out = (float*)d_out;           // same shape, f32 spikes
    (void)d_ws; (void)ws_size; (void)n_in;

    const int total     = in_sizes[0];
    const int n_spatial = total / T_STEPS;     // 2,097,152
    const int blocks    = (n_spatial + TILE_FLOATS - 1) / TILE_FLOATS; // 2048

    lif_scan_async<<<blocks, TPB, 0, stream>>>(x, out, n_spatial);
}